// CFPGv2ExplModule_47485158425174
// MI455X (gfx1250) — compile-verified
//
#include <hip/hip_runtime.h>
#include <math.h>

// CDNA5 / gfx1250, wave32. Edge-MLP uses V_WMMA_F32_16X16X4_F32.

typedef __attribute__((ext_vector_type(2))) float v2f;
typedef __attribute__((ext_vector_type(8))) float v8f;

#define ENC_H   20
#define IN_F    128
#define LATENT  60
#define DEC_H   64

// ---------------- GCN conv helper kernels ----------------

__global__ void init_kernel(float* acc, float* deg, int N) {
    int i = blockIdx.x * blockDim.x + threadIdx.x;
    if (i < N * ENC_H) acc[i] = 0.0f;
    if (i < N) deg[i] = 1.0f;   // self-loop contribution to degree
}

__global__ void deg_kernel(const long long* ei, float* deg, int E) {
    int e = blockIdx.x * blockDim.x + threadIdx.x;
    if (e < E) {
        long long c = ei[(long long)E + e];   // col = target
        atomicAdd(&deg[c], 1.0f);
    }
}

__global__ void dinv_kernel(float* deg, int N) {
    int i = blockIdx.x * blockDim.x + threadIdx.x;
    if (i < N) deg[i] = rsqrtf(deg[i]);       // deg >= 1 always
}

// h[n][k] = sum_j x[n][j] * W[j][k]   (W is [128][20] row-major)
__global__ void xw_kernel(const float* __restrict__ x, const float* __restrict__ W,
                          float* __restrict__ h, int N) {
    int idx = blockIdx.x * blockDim.x + threadIdx.x;
    if (idx >= N * ENC_H) return;
    int n = idx / ENC_H;
    int k = idx % ENC_H;
    const float* xr = x + (long long)n * IN_F;
    float acc = 0.0f;
    #pragma unroll 8
    for (int j = 0; j < IN_F; j++) acc += xr[j] * W[j * ENC_H + k];
    h[idx] = acc;
}

// acc[col] += dinv[row]*dinv[col] * h[row]
__global__ void scatter_kernel(const long long* __restrict__ ei,
                               const float* __restrict__ dinv,
                               const float* __restrict__ h,
                               float* __restrict__ acc, int E) {
    int e = blockIdx.x * blockDim.x + threadIdx.x;
    if (e >= E) return;
    long long r = ei[e];
    long long c = ei[(long long)E + e];
    float nrm = dinv[r] * dinv[c];
    const float* hr = h + r * ENC_H;
    float* ac = acc + c * ENC_H;
    #pragma unroll
    for (int k = 0; k < ENC_H; k++) atomicAdd(&ac[k], nrm * hr[k]);
}

// h1 = relu(acc + dinv[n]^2 * h[n] (self loop) + b)
__global__ void finalize_kernel(float* __restrict__ acc, const float* __restrict__ h,
                                const float* __restrict__ dinv,
                                const float* __restrict__ bg, int N) {
    int i = blockIdx.x * blockDim.x + threadIdx.x;
    if (i >= N * ENC_H) return;
    int n = i / ENC_H;
    int k = i % ENC_H;
    float di = dinv[n];
    float v = acc[i] + di * di * h[i] + bg[k];
    acc[i] = v > 0.0f ? v : 0.0f;
}

// ---------------- Edge MLP: WMMA f32 16x16x4 ----------------
// Per wave: 16 edges. Layer1: [16 x 60] @ [60 x 64] via 4 N-chunks x 15 K-steps
// of v_wmma_f32_16x16x4_f32. Layer2 fused in-register + shfl_xor reduction.

__global__ __launch_bounds__(256) void edge_mlp_kernel(
    const long long* __restrict__ ei, const float* __restrict__ h1,
    const int* __restrict__ node_id_p, const float* __restrict__ eps,
    const float* __restrict__ W1, const float* __restrict__ b1,
    const float* __restrict__ W2, const float* __restrict__ b2,
    float* __restrict__ out, int E)
{
    __shared__ float zt[8][16][65];    // per-wave z tile, padded stride 65
    __shared__ float W1s[LATENT][65];  // [k][n], padded
    __shared__ float W2s[DEC_H];
    __shared__ float b1s[DEC_H];
    __shared__ float nodeh[ENC_H];

    int tid = threadIdx.x;

    // Stage weights
    for (int i = tid; i < LATENT * DEC_H; i += 256) W1s[i >> 6][i & 63] = W1[i];
    if (tid < DEC_H) { W2s[tid] = W2[tid]; b1s[tid] = b1[tid]; }
    if (tid < ENC_H) nodeh[tid] = h1[(long long)node_id_p[0] * ENC_H + tid];
    __syncthreads();

    int lane  = tid & 31;
    int wv    = tid >> 5;
    int m     = lane & 15;     // M index (edge within tile) / N index for B
    int half  = lane >> 4;     // K-pair select within fragment
    int tile0 = blockIdx.x * 128 + wv * 16;

    // Gather z tile: [0..19]=h1[row], [20..39]=h1[col], [40..59]=h1[node_id]
    int e0 = tile0 + m;
    if (e0 < E) {
        if (half == 0) {
            long long r = ei[e0];
            const float* hr = h1 + r * ENC_H;
            #pragma unroll
            for (int k = 0; k < ENC_H; k++) zt[wv][m][k] = hr[k];
            #pragma unroll
            for (int k = 0; k < ENC_H; k++) zt[wv][m][40 + k] = nodeh[k];
        } else {
            long long c = ei[(long long)E + e0];
            const float* hc = h1 + c * ENC_H;
            #pragma unroll
            for (int k = 0; k < ENC_H; k++) zt[wv][m][20 + k] = hc[k];
        }
    }
    __syncthreads();

    // Layer 1 (WMMA) + fused layer 2 partials.
    float wacc[8] = {0.f, 0.f, 0.f, 0.f, 0.f, 0.f, 0.f, 0.f};
    #pragma unroll
    for (int nc = 0; nc < 4; nc++) {
        v8f c = {};
        int n0 = nc * 16;
        #pragma unroll
        for (int kc = 0; kc < 15; kc++) {
            int kb = kc * 4 + 2 * half;
            v2f a, b;
            // A 16x4 f32: lanes 0-15 -> K={0,1}, lanes 16-31 -> K={2,3}; M=lane%16
            a.x = zt[wv][m][kb];
            a.y = zt[wv][m][kb + 1];
            // B 4x16 f32: lanes 0-15 -> K={0,1}, lanes 16-31 -> K={2,3}; N=lane%16
            b.x = W1s[kb][n0 + m];
            b.y = W1s[kb + 1][n0 + m];
            c = __builtin_amdgcn_wmma_f32_16x16x4_f32(
                    false, a, false, b, (short)0, c, false, false);
        }
        // C layout: VGPR r, lanes 0-15 -> M=r; lanes 16-31 -> M=r+8; N=lane%16
        int n   = n0 + m;
        float bb = b1s[n];
        float w2 = W2s[n];
        #pragma unroll
        for (int r = 0; r < 8; r++) {
            float t = c[r] + bb;
            t = t > 0.0f ? t : 0.0f;
            wacc[r] += t * w2;   // layer-2 partial over this lane's N
        }
    }

    // Sum partials across the 16 lanes of each half (xor masks stay in-half).
    #pragma unroll
    for (int s = 1; s < 16; s <<= 1) {
        #pragma unroll
        for (int r = 0; r < 8; r++)
            wacc[r] += __shfl_xor(wacc[r], s, 32);
    }

    // Lanes 0 and 16 own M=0..7 and M=8..15 respectively.
    if (m == 0) {
        float b2v = b2[0];
        #pragma unroll
        for (int r = 0; r < 8; r++) {
            int mm = half * 8 + r;
            int e  = tile0 + mm;
            if (e < E) {
                float w   = wacc[r] + b2v;
                float ev  = eps[e];
                // e' = (bias - (1-bias))*eps + (1-bias), bias = 1e-4
                float eog  = 0.9999f - 0.9998f * ev;
                float gate = logf(eog) - logf(1.0f - eog) + w;
                out[e] = 1.0f / (1.0f + expf(-gate));
            }
        }
    }
}

// ---------------- launcher ----------------

extern "C" void kernel_launch(void* const* d_in, const int* in_sizes, int n_in,
                              void* d_out, int out_size, void* d_ws, size_t ws_size,
                              hipStream_t stream) {
    (void)n_in; (void)ws_size;
    const float*     x       = (const float*)d_in[0];
    const long long* ei      = (const long long*)d_in[1];
    const int*       node_id = (const int*)d_in[2];   // low word valid for i32/i64
    const float*     eps     = (const float*)d_in[3];
    const float*     Wg      = (const float*)d_in[4];
    const float*     bg      = (const float*)d_in[5];
    const float*     W1      = (const float*)d_in[6];
    const float*     b1      = (const float*)d_in[7];
    const float*     W2      = (const float*)d_in[8];
    const float*     b2      = (const float*)d_in[9];
    float*           out     = (float*)d_out;

    int N = in_sizes[0] / IN_F;        // 50000
    int E = in_sizes[3];               // 1600000 (eps count)
    (void)out_size;

    // workspace: deg/dinv [N] | h [N*20] | acc/h1 [N*20]
    float* deg = (float*)d_ws;
    float* h   = deg + N;
    float* acc = h + (size_t)N * ENC_H;

    const int B = 256;
    int gNK = (N * ENC_H + B - 1) / B;
    int gE  = (E + B - 1) / B;
    int gN  = (N + B - 1) / B;

    init_kernel<<<gNK, B, 0, stream>>>(acc, deg, N);
    deg_kernel<<<gE, B, 0, stream>>>(ei, deg, E);
    dinv_kernel<<<gN, B, 0, stream>>>(deg, N);
    xw_kernel<<<gNK, B, 0, stream>>>(x, Wg, h, N);
    scatter_kernel<<<gE, B, 0, stream>>>(ei, deg, h, acc, E);
    finalize_kernel<<<gNK, B, 0, stream>>>(acc, h, deg, bg, N);

    int gMLP = (E + 127) / 128;        // 128 edges / block (8 waves x 16)
    edge_mlp_kernel<<<gMLP, B, 0, stream>>>(ei, acc, node_id, eps,
                                            W1, b1, W2, b2, out, E);
}